// PaDiM_22703197126852
// MI455X (gfx1250) — compile-verified
//
#include <hip/hip_runtime.h>

// PaDiM running mean/covariance update for MI455X (gfx1250, wave32).
// B=32, C=192, P=3136.  Per patch: G = X X^T via V_WMMA_F32_16X16X4_F32,
// finalization fused into the covariance write (memory-bound kernel, ~1 GB
// of HBM traffic -> ~43 us floor at 23.3 TB/s).

#define BB   32      // batch
#define CC   192     // channels
#define PP   3136    // patches
#define XS   34      // LDS row stride (floats): even (8B-aligned b64 loads),
                     // 34*i mod 64 distinct for i=0..15 -> conflict-free
#define PB   2       // patches per block
#define EPSV 0.01f

typedef __attribute__((ext_vector_type(2))) float v2f;
typedef __attribute__((ext_vector_type(8))) float v8f;

__launch_bounds__(256, 2)
__global__ void padim_update_kernel(const float* __restrict__ emb,     // [B,C,P]
                                    const float* __restrict__ means,   // [P,C]
                                    const float* __restrict__ covs,    // [P,C,C]
                                    const int*   __restrict__ n_ptr,   // scalar
                                    float* __restrict__ out_means,     // [P,C]
                                    float* __restrict__ out_covs) {    // [P,C,C]
  __shared__ float ldsX[PB][CC * XS];   // X_p[c][b], padded rows
  __shared__ float ldsM[PB][CC];        // learned means per patch

  const int tid = threadIdx.x;
  const int p0  = blockIdx.x * PB;      // even -> float2 staging loads 8B aligned

  const int   n_new  = n_ptr[0] + BB;   // 96 + 32 = 128
  const float fn     = (float)n_new;
  const float inv_n  = 1.0f / fn;
  const float inv_n1 = 1.0f / (float)(n_new - 1);

  // ---- Stage X_p[c][b] = emb[b][c][p] into LDS for PB consecutive patches.
  // emb strides: b: C*P, c: P, p: 1.  One float2 covers both patches.
  for (int idx = tid; idx < CC * BB; idx += 256) {
    const int c = idx / BB;
    const int b = idx % BB;
    const float2 v = *(const float2*)(emb + (size_t)b * (CC * PP) + (size_t)c * PP + p0);
    ldsX[0][c * XS + b] = v.x;
    ldsX[1][c * XS + b] = v.y;
  }
  __syncthreads();

  // ---- Learned means: lm[c] = (means[p,c] + sum_b X[c,b]) / n_new
  for (int idx = tid; idx < PB * CC; idx += 256) {
    const int pl = idx / CC;
    const int c  = idx % CC;
    float s = 0.0f;
#pragma unroll
    for (int b = 0; b < BB; ++b) s += ldsX[pl][c * XS + b];
    const float lm = (means[(size_t)(p0 + pl) * CC + c] + s) * inv_n;
    ldsM[pl][c] = lm;
    out_means[(size_t)(p0 + pl) * CC + c] = lm;
  }
  __syncthreads();

  // ---- Per-patch Gram via fp32 WMMA: 12x12 grid of 16x16 tiles, K = 32.
  const int lane = tid & 31;
  const int wave = tid >> 5;
  const int pl   = wave >> 2;           // patch within block (0..1)
  const int wip  = wave & 3;            // wave within patch  (0..3)
  const int half = lane >> 4;           // 0: lanes 0-15, 1: lanes 16-31
  const int l16  = lane & 15;

  const float* X  = &ldsX[pl][0];
  const float* LM = &ldsM[pl][0];
  const size_t pbase = (size_t)(p0 + pl) * (CC * CC);
  const float* __restrict__ cin  = covs     + pbase;
  float*       __restrict__ cout = out_covs + pbase;

  for (int t = wip * 36; t < wip * 36 + 36; ++t) {
    const int ti = t / 12;
    const int tj = t % 12;
    const int m  = ti * 16 + l16;       // A row (output row set)
    const int nn = tj * 16 + l16;       // B col (output col set)

    // Warm the incoming covariance tile while the WMMAs run.
    __builtin_prefetch(cin + (size_t)(ti * 16 + half * 8) * CC + tj * 16 + l16, 0, 3);

    v8f acc = {};
#pragma unroll
    for (int kb = 0; kb < 8; ++kb) {
      // fp32 16x16x4 A layout: lane L holds (M=L%16, K=kb*4 + 2*(L/16) .. +1)
      const int k = kb * 4 + 2 * half;
      const v2f a = *(const v2f*)&X[m  * XS + k];   // A[m, k..k+1]
      const v2f b = *(const v2f*)&X[nn * XS + k];   // B[k..k+1, n] = X[n, k..k+1]
      acc = __builtin_amdgcn_wmma_f32_16x16x4_f32(
          /*neg_a=*/false, a, /*neg_b=*/false, b,
          /*c_mod=*/(short)0, acc, /*reuse_a=*/false, /*reuse_b=*/false);
    }

    // D layout: VGPR r, lane L -> M = r + 8*(L/16), N = L%16.
    const float lmn = LM[tj * 16 + l16];
#pragma unroll
    for (int r = 0; r < 8; ++r) {
      const int ci = ti * 16 + r + 8 * half;
      const int di = tj * 16 + l16;
      const size_t o = (size_t)ci * CC + di;
      float v = (cin[o] + acc[r] - fn * LM[ci] * lmn) * inv_n1;
      if (ci == di) v += EPSV;
      cout[o] = v;
    }
  }
}

extern "C" void kernel_launch(void* const* d_in, const int* in_sizes, int n_in,
                              void* d_out, int out_size, void* d_ws, size_t ws_size,
                              hipStream_t stream) {
  (void)in_sizes; (void)n_in; (void)out_size; (void)d_ws; (void)ws_size;
  const float* emb   = (const float*)d_in[0];   // [B,C,P]
  const float* means = (const float*)d_in[1];   // [P,C]
  const float* covs  = (const float*)d_in[2];   // [P,C,C]
  const int*   n_ptr = (const int*)d_in[3];     // scalar

  float* out_means = (float*)d_out;                         // [P,C]
  float* out_covs  = (float*)d_out + (size_t)PP * CC;       // [P,C,C]

  padim_update_kernel<<<PP / PB, 256, 0, stream>>>(emb, means, covs, n_ptr,
                                                   out_means, out_covs);
}